// GraphConvolutionalNetwork_55740085567810
// MI455X (gfx1250) — compile-verified
//
#include <hip/hip_runtime.h>
#include <hip/hip_bf16.h>
#include <math.h>

// ---------------------------------------------------------------------------
// GCN forward on gfx1250:
//   deg -> rsqrt, H1 = X@W1 (WMMA f32 16x16x4), L2-resident atomic scatter,
//   bias+relu, H2 = H1r@W2 (WMMA), scatter, bias -> d_out[h],
//   P = h@Wp1 (WMMA), head -> d_out[node_weights].
// ---------------------------------------------------------------------------

typedef __attribute__((ext_vector_type(2))) float v2f;
typedef __attribute__((ext_vector_type(8))) float v8f;

static inline int cdiv_i(long long a, long long b) { return (int)((a + b - 1) / b); }

// ---------------- degree kernels ----------------
__global__ void deg_init_k(float* __restrict__ dn, int n) {
  int i = blockIdx.x * blockDim.x + threadIdx.x;
  if (i < n) dn[i] = 1.0f;  // self-loop contributes 1 to every node's degree
}

__global__ void deg_count_k(const int* __restrict__ dst, float* __restrict__ dn, int e) {
  int i = blockIdx.x * blockDim.x + threadIdx.x;
  if (i < e) atomicAdd(&dn[dst[i]], 1.0f);
}

__global__ void deg_rsqrt_k(float* __restrict__ dn, int n) {
  int i = blockIdx.x * blockDim.x + threadIdx.x;
  if (i < n) dn[i] = rsqrtf(dn[i]);  // deg >= 1 always (self-loops)
}

// ---------------- WMMA fp32 GEMM: C[M,N] = A[M,K] @ W[K,N] ----------------
// One wave computes one 16x16 output tile via V_WMMA_F32_16X16X4_F32.
// Block = 256 threads = 8 waves. NT = N/16 tiles in N, MT = 8/NT tiles in M.
// A-frag (16x4, MxK): lanes 0-15 & 16-31 both hold M=lane&15; lane-half selects
//   the k-pair (k+0,k+1) vs (k+2,k+3) -> one 64-bit load per lane per k-step.
// B-frag (4x16, KxN): mirrored; row K striped across lanes (N = lane&15).
// C/D: VGPR r holds M = r + 8*(lane>=16), N = lane&15.
template <int K, int N>
__global__ __launch_bounds__(256) void gemm_wmma_k(const float* __restrict__ A,
                                                   const float* __restrict__ W,
                                                   float* __restrict__ C, int M) {
  constexpr int NT = N / 16;
  constexpr int MT = 8 / NT;
  constexpr bool USE_LDS = (K * N * 4) <= 32768;  // stage small weights in LDS
  __shared__ float sW[USE_LDS ? (K * N) : 1];
  if constexpr (USE_LDS) {
    for (int i = threadIdx.x; i < K * N; i += 256) sW[i] = W[i];
    __syncthreads();
  }

  const int wave = threadIdx.x >> 5;
  const int lane = threadIdx.x & 31;
  const int l16  = lane & 15;
  const int hi   = lane >> 4;  // 0: k+0/k+1, 1: k+2/k+3
  const int mTile = blockIdx.x * MT + wave / NT;
  const int nTile = wave % NT;
  if (mTile * 16 >= M) return;  // wave-uniform: EXEC stays all-ones for WMMA
  const int nBase = nTile * 16;

  const float* aP = A + (size_t)(mTile * 16 + l16) * K + 2 * hi;

  v8f acc = {};
  for (int k = 0; k < K; k += 4) {
    v2f a, b;
    a.x = aP[k];
    a.y = aP[k + 1];
    const int kb = k + 2 * hi;
    if constexpr (USE_LDS) {
      b.x = sW[kb * N + nBase + l16];
      b.y = sW[(kb + 1) * N + nBase + l16];
    } else {
      b.x = W[(size_t)kb * N + nBase + l16];
      b.y = W[(size_t)(kb + 1) * N + nBase + l16];
    }
    acc = __builtin_amdgcn_wmma_f32_16x16x4_f32(
        /*neg_a=*/false, a, /*neg_b=*/false, b,
        /*c_mod=*/(short)0, acc, /*reuse_a=*/false, /*reuse_b=*/false);
  }

  float* out = C + (size_t)(mTile * 16 + hi * 8) * N + nBase + l16;
#pragma unroll
  for (int r = 0; r < 8; ++r) out[(size_t)r * N] = acc[r];
}

// ---------------- aggregation ----------------
// Self-loop term: agg[v] = h[v] * dn[v]^2  (also zero-initializes agg buffer)
template <int F>
__global__ void agg_init_k(const float* __restrict__ H, const float* __restrict__ dn,
                           float* __restrict__ agg, int n) {
  long long i = (long long)blockIdx.x * blockDim.x + threadIdx.x;
  if (i < (long long)n * F) {
    int v = (int)(i / F);
    float s = dn[v];
    agg[i] = H[i] * s * s;
  }
}

// Edge scatter: F/4 threads per edge, float4 gather + 4x global_atomic_add_f32.
// agg buffer (<=25.6MB) is L2-resident on MI455X (192MB L2).
template <int F>
__global__ __launch_bounds__(256) void scatter_k(const float* __restrict__ H,
                                                 const int* __restrict__ src,
                                                 const int* __restrict__ dst,
                                                 const float* __restrict__ dn,
                                                 float* __restrict__ agg, int e) {
  constexpr int TPE = F / 4;
  long long t = (long long)blockIdx.x * blockDim.x + threadIdx.x;
  int ei = (int)(t / TPE);
  int c  = (int)(t % TPE);
  if (ei >= e) return;
  int s = src[ei];
  int d = dst[ei];
  float coef = dn[s] * dn[d];
  float4 v = reinterpret_cast<const float4*>(H + (size_t)s * F)[c];
  float* ap = agg + (size_t)d * F + c * 4;
  atomicAdd(ap + 0, v.x * coef);
  atomicAdd(ap + 1, v.y * coef);
  atomicAdd(ap + 2, v.z * coef);
  atomicAdd(ap + 3, v.w * coef);
}

// ---------------- bias (+ optional relu) ----------------
template <int F, bool RELU>
__global__ void bias_k(const float* __restrict__ X, const float* __restrict__ b,
                       float* __restrict__ out, int n) {
  long long i = (long long)blockIdx.x * blockDim.x + threadIdx.x;
  if (i < (long long)n * F) {
    float v = X[i] + b[(int)(i % F)];
    out[i] = RELU ? fmaxf(v, 0.0f) : v;
  }
}

// ---------------- MLP head epilogue ----------------
// nw[v] = sigmoid( bp2 + sum_t relu(P[v,t] + bp1[t]) * Wp2[t] )
__global__ void head_k(const float* __restrict__ P, const float* __restrict__ bp1,
                       const float* __restrict__ Wp2, const float* __restrict__ bp2,
                       float* __restrict__ nw, int n) {
  int v = blockIdx.x * blockDim.x + threadIdx.x;
  if (v >= n) return;
  const float* p = P + (size_t)v * 64;
  float acc = bp2[0];
#pragma unroll 8
  for (int t = 0; t < 64; ++t) {
    float pt = fmaxf(p[t] + bp1[t], 0.0f);
    acc = fmaf(pt, Wp2[t], acc);
  }
  nw[v] = 1.0f / (1.0f + __expf(-acc));
}

// ---------------------------------------------------------------------------
extern "C" void kernel_launch(void* const* d_in, const int* in_sizes, int n_in,
                              void* d_out, int out_size, void* d_ws, size_t ws_size,
                              hipStream_t stream) {
  const float* x   = (const float*)d_in[0];
  const int*   edg = (const int*)d_in[1];
  const float* W1  = (const float*)d_in[2];
  const float* b1  = (const float*)d_in[3];
  const float* W2  = (const float*)d_in[4];
  const float* b2  = (const float*)d_in[5];
  const float* Wp1 = (const float*)d_in[6];
  const float* bp1 = (const float*)d_in[7];
  const float* Wp2 = (const float*)d_in[8];
  const float* bp2 = (const float*)d_in[9];

  const int n = in_sizes[0] / 256;  // N_NODES = 100000 (multiple of 16)
  const int e = in_sizes[1] / 2;    // N_EDGES = 3200000
  const int* src = edg;
  const int* dst = edg + e;

  // workspace layout (floats): dn[n] | bufA[n*64] | bufB[n*64]
  float* dn   = (float*)d_ws;
  float* bufA = dn + n;
  float* bufB = bufA + (size_t)n * 64;

  float* outH = (float*)d_out;           // [n,32] final conv output
  float* outW = outH + (size_t)n * 32;   // [n,1]  node weights

  const int B = 256;
  const int mtiles = n / 16;

  // degree normalization
  deg_init_k<<<cdiv_i(n, B), B, 0, stream>>>(dn, n);
  deg_count_k<<<cdiv_i(e, B), B, 0, stream>>>(dst, dn, e);
  deg_rsqrt_k<<<cdiv_i(n, B), B, 0, stream>>>(dn, n);

  // ---- GCN layer 1: H1 = X @ W1 ; agg ; +b1 ; relu ----
  gemm_wmma_k<256, 64><<<cdiv_i(mtiles, 2), B, 0, stream>>>(x, W1, bufA, n);
  agg_init_k<64><<<cdiv_i((long long)n * 64, B), B, 0, stream>>>(bufA, dn, bufB, n);
  scatter_k<64><<<cdiv_i((long long)e * 16, B), B, 0, stream>>>(bufA, src, dst, dn, bufB, e);
  bias_k<64, true><<<cdiv_i((long long)n * 64, B), B, 0, stream>>>(bufB, b1, bufB, n);

  // ---- GCN layer 2: H2 = H1r @ W2 ; agg ; +b2 -> d_out ----
  gemm_wmma_k<64, 32><<<cdiv_i(mtiles, 4), B, 0, stream>>>(bufB, W2, bufA, n);
  agg_init_k<32><<<cdiv_i((long long)n * 32, B), B, 0, stream>>>(bufA, dn, bufB, n);
  scatter_k<32><<<cdiv_i((long long)e * 8, B), B, 0, stream>>>(bufA, src, dst, dn, bufB, e);
  bias_k<32, false><<<cdiv_i((long long)n * 32, B), B, 0, stream>>>(bufB, b2, outH, n);

  // ---- weight predictor: P = h @ Wp1 ; head -> sigmoid ----
  gemm_wmma_k<32, 64><<<cdiv_i(mtiles, 2), B, 0, stream>>>(outH, Wp1, bufA, n);
  head_k<<<cdiv_i(n, B), B, 0, stream>>>(bufA, bp1, Wp2, bp2, outW, n);
}